// TorchCombineModule_47880295416400
// MI455X (gfx1250) — compile-verified
//
#include <hip/hip_runtime.h>
#include <hip/hip_bf16.h>
#include <stdint.h>

// ---------------------------------------------------------------------------
// MoE combine (gather-permute-copy with fp32 -> bf16 conversion)
//
// dispatched_buffer: (G=1, C=8, E=8, CAP=2048, H=1024) fp32
// metadata:          (G, C, E, CAP, 3) int32  [meta0, token, topk]
// expert_token_counts: (G, C, E) int32  (== 1024 here)
// out:               (C, S=1024, K=8, H=1024) bf16
//
// Valid entries form a permutation of all C*S*K destination rows, so every
// output row is written exactly once -> plain stores, no zero-init, no atomics.
//
// Roofline: 256 MB read + 128 MB write = 384 MB @ 23.3 TB/s ~= 16.5 us floor.
// Pure streaming data movement:
//   * CDNA5 async global->LDS DMA (global_load_async_to_lds_b128, ASYNCcnt)
//     stages 4 KB rows; 4 rows in flight per block before one wait.
//   * Lane-private LDS flow (each lane reads back only the 32 B it DMA'd)
//     -> no workgroup barrier needed after s_wait_asynccnt.
//   * b128 stores (8 bf16 per lane), b128 async loads (16 B x2 per lane).
//   * Touch-once streams -> th:TH_LOAD_NT / th:TH_STORE_NT to bypass L2
//     retention (192 MB L2 has zero reuse to exploit here).
// ---------------------------------------------------------------------------

namespace cfg {
constexpr int G   = 1;
constexpr int C   = 8;
constexpr int E   = 8;
constexpr int CAP = 2048;
constexpr int S   = 1024;
constexpr int K   = 8;
constexpr int H   = 1024;
constexpr int RP      = 4;     // rows staged per block (16 KB LDS)
constexpr int THREADS = 256;   // 8 wave32; 128 lanes per row, 2 rows at once
}

typedef __attribute__((ext_vector_type(4))) unsigned int v4u;

// fp32 -> bf16 bits, round-to-nearest-even (finite inputs; matches jnp astype)
__device__ __forceinline__ uint32_t f32_to_bf16_rne(float f) {
    uint32_t u = __float_as_uint(f);
    u += 0x7FFFu + ((u >> 16) & 1u);
    return u >> 16;
}
__device__ __forceinline__ uint32_t pack_bf16x2(float lo, float hi) {
    return f32_to_bf16_rne(lo) | (f32_to_bf16_rne(hi) << 16);
}

__global__ __launch_bounds__(cfg::THREADS)
void moe_combine_kernel(const float* __restrict__ buf,
                        const int*   __restrict__ meta,
                        const int*   __restrict__ counts,
                        unsigned short* __restrict__ out) {
    using namespace cfg;

    // 4 row buffers of 4 KB in LDS (16 KB total; WGP has 320 KB).
    __shared__ float lds_row[RP][H];

    const int tid  = threadIdx.x;
    const int half = tid >> 7;     // which of the 2 concurrent rows
    const int lane = tid & 127;    // 128 lanes x 32 B == one 4 KB row
    const int row0 = blockIdx.x * RP;

    // ---- Decode metadata for all RP rows (blockIdx-uniform -> s_load path).
    int  dst_row[RP];
    bool valid[RP];
    #pragma unroll
    for (int r = 0; r < RP; ++r) {
        const int row  = row0 + r;                 // flat (g,c,e,slot)
        const int slot = row % CAP;
        const int e    = (row / CAP) % E;
        const int c    = (row / (CAP * E)) % C;
        const int g    =  row / (CAP * E * C);

        const int cnt = counts[(g * C + c) * E + e];
        const int* m  = meta + (size_t)row * 3;
        const int m0  = m[0];
        const int tok = m[1];
        const int tk  = m[2];

        valid[r]   = (slot < cnt) && ((m0 % G) == g);
        const int src_chip = m0 / G;
        dst_row[r] = (src_chip * S + tok) * K + tk;   // flat (C,S,K) row index
    }

    // ---- Phase 1: issue async DMA. Each lane moves its own 32 B via two
    // b128 transfers; offset:16 shifts BOTH global and LDS addresses (ISA).
    #pragma unroll
    for (int i = 0; i < RP / 2; ++i) {
        const int r = 2 * i + half;
        if (valid[r]) {
            const float* gp = buf + (size_t)(row0 + r) * H + (size_t)lane * 8;
            uint32_t la = (uint32_t)(uintptr_t)(&lds_row[r][lane * 8]); // LDS byte offset
            uint64_t ga = (uint64_t)(uintptr_t)gp;
            asm volatile("global_load_async_to_lds_b128 %0, %1, off th:TH_LOAD_NT"
                         :: "v"(la), "v"(ga) : "memory");
            asm volatile("global_load_async_to_lds_b128 %0, %1, off offset:16 th:TH_LOAD_NT"
                         :: "v"(la), "v"(ga) : "memory");
        }
    }

    // ---- Phase 2: wait for this wave's async transfers (ASYNCcnt -> 0).
    asm volatile("s_wait_asynccnt 0x0" ::: "memory");

    // ---- Phase 3: read own 32 B back, convert 8x fp32 -> bf16, b128 NT store.
    #pragma unroll
    for (int i = 0; i < RP / 2; ++i) {
        const int r = 2 * i + half;
        if (!valid[r]) continue;
        const float4* lp = reinterpret_cast<const float4*>(&lds_row[r][lane * 8]);
        const float4 a = lp[0];
        const float4 b = lp[1];
        v4u o;
        o.x = pack_bf16x2(a.x, a.y);
        o.y = pack_bf16x2(a.z, a.w);
        o.z = pack_bf16x2(b.x, b.y);
        o.w = pack_bf16x2(b.z, b.w);
        uint64_t da = (uint64_t)(uintptr_t)(out + (size_t)dst_row[r] * H
                                                + (size_t)lane * 8);
        asm volatile("global_store_b128 %0, %1, off th:TH_STORE_NT"
                     :: "v"(da), "v"(o) : "memory");
    }
}

extern "C" void kernel_launch(void* const* d_in, const int* in_sizes, int n_in,
                              void* d_out, int out_size, void* d_ws, size_t ws_size,
                              hipStream_t stream) {
    using namespace cfg;
    (void)in_sizes; (void)n_in; (void)out_size; (void)d_ws; (void)ws_size;

    const float* buf      = (const float*)d_in[0];
    const int*   meta     = (const int*)d_in[1];
    const int*   counts   = (const int*)d_in[2];
    unsigned short* out   = (unsigned short*)d_out;   // bf16 bit pattern

    const int total_rows = G * C * E * CAP;           // 131072
    const int blocks     = total_rows / RP;           // 32768

    moe_combine_kernel<<<dim3(blocks), dim3(THREADS), 0, stream>>>(
        buf, meta, counts, out);
}